// SRLL_33483565039957
// MI455X (gfx1250) — compile-verified
//
#include <hip/hip_runtime.h>
#include <hip/hip_bf16.h>
#include <math.h>

// Problem constants from the reference: N=256 images, H=240, W=320, P=256 ids.
#define NIMG 256
#define HW   (240 * 320)
#define NP   256

// ---------------------------------------------------------------------------
// CDNA5 async global->LDS gather (ASYNCcnt-tracked). Builtin if the toolchain
// declares it, otherwise inline asm per CDNA5 ISA.
// The builtin's params are int* in addrspace(1) (global) and addrspace(3) (LDS)
// per the round-1 diagnostic.
// ---------------------------------------------------------------------------
#if defined(__has_builtin)
#  if __has_builtin(__builtin_amdgcn_global_load_async_to_lds_b32)
#    define HAVE_ASYNC_BUILTIN 1
#  endif
#  if __has_builtin(__builtin_amdgcn_s_wait_asynccnt)
#    define HAVE_WAIT_ASYNC_BUILTIN 1
#  endif
#endif

typedef __attribute__((address_space(1))) int gint_as1;
typedef __attribute__((address_space(3))) int lint_as3;

__device__ __forceinline__ void async_gather_b32(const float* gsrc, float* lds_dst) {
#if defined(HAVE_ASYNC_BUILTIN)
    __builtin_amdgcn_global_load_async_to_lds_b32(
        (gint_as1*)gsrc, (lint_as3*)lds_dst, /*offset=*/0, /*cpol=*/0);
#else
    // VDST VGPR = per-lane LDS byte address (low 32 bits of the generic
    // pointer to an LDS object are the LDS offset), VADDR = 64-bit global addr.
    unsigned int lds_addr = (unsigned int)(unsigned long long)(size_t)lds_dst;
    asm volatile("global_load_async_to_lds_b32 %0, %1, off"
                 :: "v"(lds_addr), "v"(gsrc)
                 : "memory");
#endif
}

__device__ __forceinline__ void wait_async_zero() {
#if defined(HAVE_WAIT_ASYNC_BUILTIN)
    __builtin_amdgcn_s_wait_asynccnt(0);
#else
    asm volatile("s_wait_asynccnt 0" ::: "memory");
#endif
    // Compiler-level memory fence: the async engine wrote LDS behind the
    // compiler's back; force it to treat LDS as modified.
    asm volatile("" ::: "memory");
}

// ---------------------------------------------------------------------------
// Kernel 1: one block per image. Gather P points into LDS, evaluate all P*P
// pairs, tree-reduce, emit one (num, den) double pair per block.
// ---------------------------------------------------------------------------
__global__ __launch_bounds__(NP) void srll_pair_kernel(
    const float* __restrict__ fake, const float* __restrict__ real,
    const float* __restrict__ mask, const int* __restrict__ ids,
    double* __restrict__ part) {
    __shared__ float sfi[NP];
    __shared__ float srv[NP];
    __shared__ float smi[NP];
    __shared__ double rnum[NP];
    __shared__ double rden[NP];

    const int t = threadIdx.x;
    const int n = blockIdx.x;
    const long long base = (long long)n * HW + (long long)ids[t];

    // Scattered gather straight into LDS via the async path.
    async_gather_b32(fake + base, &sfi[t]);
    async_gather_b32(real + base, &srv[t]);
    async_gather_b32(mask + base, &smi[t]);
    wait_async_zero();
    __syncthreads();

    const float fi = sfi[t];
    const float ri = srv[t];
    const float mi = smi[t];

    double anum = 0.0;
    double aden = 0.0;
#pragma unroll 4
    for (int j = 0; j < NP; ++j) {
        const float fj = sfi[j];
        const float rj = srv[j];
        const float m  = mi * smi[j];
        const float d  = fi - fj;
        const bool  b1 = ri > 1.02f * rj;       // "real_i beats real_j"
        const bool  b2 = rj > 1.02f * ri;       // "real_j beats real_i"
        // r1 branch: loss=log1p(e^{-d}), w=sigmoid(-d)^2  -> s=-d
        // r2 branch: loss=log1p(e^{ d}), w=sigmoid( d)^2  -> s=+d
        // r0 branch: loss=d^2, w=(1-sigmoid(0))^2=0.25
        const float s   = b1 ? -d : d;
        const float e   = expf(s);
        const float sig = e / (1.0f + e);       // sigmoid(s)
        const float rank_val = log1pf(e) * sig * sig;
        const float val = (b1 || b2) ? rank_val : 0.25f * d * d;
        anum += (double)(m * val);
        aden += (double)m;
    }

    rnum[t] = anum;
    rden[t] = aden;
    __syncthreads();
#pragma unroll
    for (int s = NP / 2; s > 0; s >>= 1) {
        if (t < s) {
            rnum[t] += rnum[t + s];
            rden[t] += rden[t + s];
        }
        __syncthreads();
    }
    if (t == 0) {
        part[2 * n + 0] = rnum[0];
        part[2 * n + 1] = rden[0];
    }
}

// ---------------------------------------------------------------------------
// Kernel 2: deterministic final reduction of per-block partials -> scalar.
// ---------------------------------------------------------------------------
__global__ __launch_bounds__(256) void srll_finish_kernel(
    const double* __restrict__ part, float* __restrict__ out, int nblk) {
    __shared__ double rn[256];
    __shared__ double rd[256];
    const int t = threadIdx.x;
    double a = 0.0, b = 0.0;
    for (int i = t; i < nblk; i += 256) {
        a += part[2 * i + 0];
        b += part[2 * i + 1];
    }
    rn[t] = a;
    rd[t] = b;
    __syncthreads();
#pragma unroll
    for (int s = 128; s > 0; s >>= 1) {
        if (t < s) {
            rn[t] += rn[t + s];
            rd[t] += rd[t + s];
        }
        __syncthreads();
    }
    if (t == 0) {
        out[0] = (float)(rn[0] / (rd[0] + 1e-6));
    }
}

// ---------------------------------------------------------------------------
extern "C" void kernel_launch(void* const* d_in, const int* in_sizes, int n_in,
                              void* d_out, int out_size, void* d_ws, size_t ws_size,
                              hipStream_t stream) {
    const float* fake = (const float*)d_in[0];
    const float* real = (const float*)d_in[1];
    const float* mask = (const float*)d_in[2];
    const int*   ids  = (const int*)d_in[3];
    double* part = (double*)d_ws;   // 2 doubles per block = 4 KB

    srll_pair_kernel<<<NIMG, NP, 0, stream>>>(fake, real, mask, ids, part);
    srll_finish_kernel<<<1, 256, 0, stream>>>(part, (float*)d_out, NIMG);
}